// FlowMixtureNLL_2697239462528
// MI455X (gfx1250) — compile-verified
//
#include <hip/hip_runtime.h>

typedef __attribute__((ext_vector_type(2))) float v2f;
typedef __attribute__((ext_vector_type(8))) float v8f;

#define K_ 8
#define L_ 2
#define B_ 32
#define D_ 256
#define S_ 1024

// ---------------------------------------------------------------------------
// Kernel 1: per-(k,b,s) Gaussian term, D-reduction via V_WMMA_F32_16X16X4_F32.
//
// Each wave handles one (k, b, 16-wide s tile). Per d-step of 4, the wave
// forms the 4(d) x 16(s) tile of  t = lv0 + lv1 + (x-mu)^2 * exp(-lv0)
// in the B-matrix operand (column n = lane%16 = s offset; the two v2f
// elements of a lane are two different d-rows of the same column), and
// accumulates C += ones(16x4) x B(4x16), i.e. column sums over d.
// Because A is all-ones, the result is invariant to the exact row order,
// so only the column=lane%16 striping is load-bearing.
// After 64 steps, c[0] holds sum_d t for s = s0 + (lane & 15) in every lane.
// ---------------------------------------------------------------------------
__global__ __launch_bounds__(256) void pnll_kernel(
    const float* __restrict__ mus,
    const float* __restrict__ logvars,
    const float* __restrict__ samples,
    float* __restrict__ pnll) {
  const int tid  = threadIdx.x;
  const int lane = tid & 31;
  const int wave = tid >> 5;

  const int tile = blockIdx.x * 8 + wave;   // [0, K*B*(S/16)) = [0, 16384)
  const int s16  = tile & 63;               // S/16 = 64 tiles per (k,b)
  const int b    = (tile >> 6) & 31;
  const int k    = tile >> 11;

  const int s   = (s16 << 4) + (lane & 15); // this lane's s column
  const int dsh = (lane >> 4) << 1;         // lanes 0-15: d+{0,1}; 16-31: d+{2,3}

  const size_t baseKB = (size_t)(k * B_ + b) * D_ * S_;
  const float* __restrict__ mu = mus + baseKB;
  const float* __restrict__ sm = samples + baseKB;
  const float* __restrict__ l0 = logvars + (size_t)((k * L_ + 0) * B_ + b) * D_ * S_;
  const float* __restrict__ l1 = logvars + (size_t)((k * L_ + 1) * B_ + b) * D_ * S_;

  const v2f ones = {1.0f, 1.0f};
  v8f c = {};

#pragma unroll 4
  for (int d0 = 0; d0 < D_; d0 += 4) {
    const int o0 = (d0 + dsh) * S_ + s;  // row d0+dsh   of this column
    const int o1 = o0 + S_;              // row d0+dsh+1 of this column

    const float m0 = mu[o0], m1 = mu[o1];
    const float x0 = sm[o0], x1 = sm[o1];
    const float a0 = l0[o0], a1 = l0[o1];
    const float g0 = l1[o0], g1 = l1[o1];

    const float e0 = x0 - m0;
    const float e1 = x1 - m1;
    const float t0 = a0 + g0 + e0 * e0 * __expf(-a0);
    const float t1 = a1 + g1 + e1 * e1 * __expf(-a1);

    v2f bm = {t0, t1};
    // D = A(16x4 ones) * B(4x16 t-tile) + C  -> column sums over d
    c = __builtin_amdgcn_wmma_f32_16x16x4_f32(
        /*neg_a=*/false, ones, /*neg_b=*/false, bm,
        /*c_mod=*/(short)0, c, /*reuse_a=*/false, /*reuse_b=*/false);
  }

  // c[0]: lanes 0-15 hold D[0, n=lane]; lanes 16-31 hold D[8, n=lane-16];
  // all rows equal -> every lane has its column's sum. Lanes 0-15 store.
  if (lane < 16) {
    constexpr float HALF_LOG2PI_D = 0.5f * 1.8378770664093453f * (float)D_;
    const float pn = -0.5f * c[0] - HALF_LOG2PI_D;
    pnll[(size_t)(k * B_ + b) * S_ + s] = pn;
  }
}

// ---------------------------------------------------------------------------
// Kernel 2: per-(b,s) logsumexp over K with log mixture weights, then
// block-level reduction of -lp/B into the scalar output.
// ---------------------------------------------------------------------------
__global__ __launch_bounds__(256) void nll_reduce_kernel(
    const float* __restrict__ pnll,
    const float* __restrict__ logits,
    float* __restrict__ out) {
  const int idx = blockIdx.x * 256 + threadIdx.x;  // [0, B*S)
  const int s = idx & (S_ - 1);
  const int b = idx >> 10;

  // log-softmax of logits[b, :] (8 values, L2-resident)
  float lg[K_];
  float mw = -3.0e38f;
  for (int kk = 0; kk < K_; ++kk) {
    lg[kk] = logits[b * K_ + kk];
    mw = fmaxf(mw, lg[kk]);
  }
  float sw = 0.0f;
  for (int kk = 0; kk < K_; ++kk) sw += __expf(lg[kk] - mw);
  const float lse_w = mw + __logf(sw);

  // LSE over K of pnll + log_w
  float v[K_];
  float mv = -3.0e38f;
  for (int kk = 0; kk < K_; ++kk) {
    v[kk] = pnll[(size_t)(kk * B_ + b) * S_ + s] + (lg[kk] - lse_w);
    mv = fmaxf(mv, v[kk]);
  }
  float sv = 0.0f;
  for (int kk = 0; kk < K_; ++kk) sv += __expf(v[kk] - mv);
  const float lp = mv + __logf(sv);

  float contrib = -lp * (1.0f / (float)B_);

  // wave32 reduction
  for (int off = 16; off > 0; off >>= 1)
    contrib += __shfl_xor(contrib, off, 32);

  __shared__ float wsum[8];
  if ((threadIdx.x & 31) == 0) wsum[threadIdx.x >> 5] = contrib;
  __syncthreads();
  if (threadIdx.x == 0) {
    float t = 0.0f;
    for (int i = 0; i < 8; ++i) t += wsum[i];
    atomicAdd(out, t);
  }
}

__global__ void zero_kernel(float* __restrict__ out) { out[0] = 0.0f; }

extern "C" void kernel_launch(void* const* d_in, const int* in_sizes, int n_in,
                              void* d_out, int out_size, void* d_ws, size_t ws_size,
                              hipStream_t stream) {
  (void)in_sizes; (void)n_in; (void)out_size; (void)ws_size;
  const float* mus     = (const float*)d_in[0];
  const float* logvars = (const float*)d_in[1];
  const float* samples = (const float*)d_in[2];
  const float* logits  = (const float*)d_in[3];
  float* out  = (float*)d_out;
  float* pnll = (float*)d_ws;  // K*B*S floats = 1 MiB scratch

  zero_kernel<<<1, 1, 0, stream>>>(out);

  // K*B*(S/16) = 16384 wave-tiles / 8 waves per block = 2048 blocks
  pnll_kernel<<<(K_ * B_ * (S_ / 16)) / 8, 256, 0, stream>>>(mus, logvars, samples, pnll);

  // B*S = 32768 sites / 256 threads = 128 blocks
  nll_reduce_kernel<<<(B_ * S_) / 256, 256, 0, stream>>>(pnll, logits, out);
}